// aveSupConLoss_26087631356315
// MI455X (gfx1250) — compile-verified
//
#include <hip/hip_runtime.h>
#include <hip/hip_bf16.h>

// ---------------------------------------------------------------------------
// SupCon loss, MI455X (gfx1250, wave32).
//   X = contrast = [4096, 256] f32   (from features [2048, 2, 256])
//   G = X @ X^T  computed 16-row strips at a time, strip lives ONLY in LDS
//   (256 KB strip + 16 KB class table + cnt table < 320 KB WGP LDS).
//   GEMM uses V_WMMA_F32_16X16X4_F32 (f32 in / f32 acc).
//   B loads are chunked (8 at a time) and two independent accumulator chains
//   run per wave so WMMA issue overlaps L2 latency instead of serializing.
// ---------------------------------------------------------------------------

typedef __attribute__((ext_vector_type(2))) float v2f;
typedef __attribute__((ext_vector_type(8))) float v8f;

#define NROWS      4096
#define DDIM       256
#define HALFB      2048
#define NCLS_PAD   128
#define INV_TEMP   14.2857142857142857f   // 1 / 0.07
#define STRIP_ROWS 16

// ---- kernel 1: gather contrast layout: X[i] = features[i<2048 ? (i,0) : (i-2048,1)]
__global__ void build_contrast_kernel(const float* __restrict__ feat,
                                      float* __restrict__ X) {
    int t = blockIdx.x * blockDim.x + threadIdx.x;      // 0 .. 262143 (float4 units)
    int row = t >> 6;                                   // 64 float4 per 256-float row
    int c4  = t & 63;
    int src = (row < HALFB) ? (row * 2) : ((row - HALFB) * 2 + 1);
    float4 v = *(const float4*)(feat + src * DDIM + c4 * 4);
    *(float4*)(X + row * DDIM + c4 * 4) = v;
}

// ---- kernel 2: per-column class id + per-class counts (one block, 256 thr)
__global__ void build_classes_kernel(const int* __restrict__ labels,
                                     int* __restrict__ cls,
                                     float* __restrict__ cnt) {
    int tid = threadIdx.x;
    if (tid < NCLS_PAD) cnt[tid] = 0.0f;
    __syncthreads();
    for (int j = tid; j < NROWS; j += 256) {
        int c = labels[j & (HALFB - 1)];                // targets = tile(labels, 2)
        cls[j] = c;
        atomicAdd(&cnt[c], 1.0f);
    }
}

// ---- kernel 3: fused strip GEMM (WMMA f32) + row reductions
//      grid = 256 blocks (16 rows each), block = 256 threads = 8 waves
//      279 KB LDS forces 1 block/WGP -> tell the compiler (full VGPR budget).
__global__ void __launch_bounds__(256, 1)
supcon_strip_kernel(const float* __restrict__ X,
                    const int* __restrict__ cls,
                    const float* __restrict__ cnt,
                    float* __restrict__ rowloss) {
    extern __shared__ float lds[];
    float* strip = lds;                                  // 16 * 4096 f32 = 256 KB
    int*   clsL  = (int*)(lds + STRIP_ROWS * NROWS);     // 4096 i32   =  16 KB
    float* cntL  = (float*)(clsL + NROWS);               // 128 f32    = 512 B

    const int tid  = threadIdx.x;
    const int lane = tid & 31;
    const int wave = tid >> 5;
    const int r0   = blockIdx.x * STRIP_ROWS;

    // cooperative load of class table + counts into LDS
    for (int j = tid; j < NROWS; j += 256) clsL[j] = cls[j];
    if (tid < NCLS_PAD) cntL[tid] = cnt[tid];

    // ---------------- GEMM stage ----------------
    // f32 WMMA 16x16x4 operand mapping (ISA 7.12.2):
    //   A 16x4 : lane holds M = lane&15, float2 at K = k0 + 2*(lane>>4)
    //   B  4x16: B[k][n] = X[n][k]  -> identical per-lane load pattern
    //   C/D    : VGPR r -> M = r + 8*(lane>>4), N = lane&15
    const int lm = lane & 15;
    const int lh = lane >> 4;

    // Preload this block's A strip (16 rows x 256 K) into registers: 64 x v2f
    v2f a[64];
    const float* abase = X + (r0 + lm) * DDIM + 2 * lh;
#pragma unroll
    for (int kk = 0; kk < 64; ++kk)
        a[kk] = *(const v2f*)(abase + kk * 4);

    // wave w owns N-columns [512w, 512w+512): 32 tiles of 16, taken in pairs
    const int n0 = wave * 512;
    for (int tp = 0; tp < 16; ++tp) {
        const int nb0 = n0 + tp * 32;        // tile pair: columns nb0.. , nb0+16..
        const int nb1 = nb0 + 16;
        const float* bbase0 = X + (nb0 + lm) * DDIM + 2 * lh;
        const float* bbase1 = X + (nb1 + lm) * DDIM + 2 * lh;
        if (tp + 1 < 16) {
            __builtin_prefetch(X + (nb0 + 32 + lm) * DDIM, 0, 1);
            __builtin_prefetch(X + (nb0 + 48 + lm) * DDIM, 0, 1);
        }

        v8f acc0 = {0.f, 0.f, 0.f, 0.f, 0.f, 0.f, 0.f, 0.f};
        v8f acc1 = {0.f, 0.f, 0.f, 0.f, 0.f, 0.f, 0.f, 0.f};

        // K loop in chunks of 8: issue 16 loads back-to-back, then 16 WMMAs
        // across two independent accumulator chains.
#pragma unroll
        for (int kc = 0; kc < 8; ++kc) {
            v2f b0[8], b1[8];
#pragma unroll
            for (int u = 0; u < 8; ++u) {
                const int kk = kc * 8 + u;
                b0[u] = *(const v2f*)(bbase0 + kk * 4);
                b1[u] = *(const v2f*)(bbase1 + kk * 4);
            }
#pragma unroll
            for (int u = 0; u < 8; ++u) {
                const int kk = kc * 8 + u;
                acc0 = __builtin_amdgcn_wmma_f32_16x16x4_f32(
                    false, a[kk], false, b0[u], (short)0, acc0, false, false);
                acc1 = __builtin_amdgcn_wmma_f32_16x16x4_f32(
                    false, a[kk], false, b1[u], (short)0, acc1, false, false);
            }
        }
#pragma unroll
        for (int r = 0; r < 8; ++r) {
            strip[(r + 8 * lh) * NROWS + nb0 + lm] = acc0[r];
            strip[(r + 8 * lh) * NROWS + nb1 + lm] = acc1[r];
        }
    }
    __syncthreads();

    // ---------------- reduction stage ----------------
    // wave w handles strip rows {2w, 2w+1}; wave32 shuffle reductions
    for (int rr = 0; rr < 2; ++rr) {
        const int r  = wave * 2 + rr;
        const int gi = r0 + r;                  // global row (anchor) index
        const int ci = clsL[gi];
        const float* row = strip + r * NROWS;

        // pass 1: row max of raw dot products (max of logits = max(g)/T)
        float m = -3.402823466e38f;
        for (int j = lane; j < NROWS; j += 32)
            m = fmaxf(m, row[j]);
#pragma unroll
        for (int off = 16; off > 0; off >>= 1)
            m = fmaxf(m, __shfl_xor(m, off, 32));

        // pass 2: S = sum_{j!=i} exp(l_ij)/w_ij ,  P = sum_j mask_ij * l_ij
        float S = 0.f, P = 0.f;
        for (int j = lane; j < NROWS; j += 32) {
            const float l = (row[j] - m) * INV_TEMP;
            const int   c = clsL[j];
            const bool mk = (c == ci) && (j != gi);
            const float w = cntL[c] - (mk ? 1.0f : 0.0f);
            if (j != gi) S += __expf(l) / w;
            if (mk)      P += l;
        }
#pragma unroll
        for (int off = 16; off > 0; off >>= 1) {
            S += __shfl_xor(S, off, 32);
            P += __shfl_xor(P, off, 32);
        }
        if (lane == 0) {
            const float Mi = cntL[ci] - 1.0f;   // >= 1 (each label appears twice)
            rowloss[gi] = (P - Mi * __logf(S)) / Mi;
        }
    }
}

// ---- kernel 4: deterministic final mean (fixed-order tree reduction)
__global__ void finalize_kernel(const float* __restrict__ rowloss,
                                float* __restrict__ out) {
    __shared__ float sc[256];
    float s = 0.f;
    for (int j = threadIdx.x; j < NROWS; j += 256) s += rowloss[j];
    sc[threadIdx.x] = s;
    __syncthreads();
    for (int off = 128; off > 0; off >>= 1) {
        if (threadIdx.x < off) sc[threadIdx.x] += sc[threadIdx.x + off];
        __syncthreads();
    }
    if (threadIdx.x == 0) out[0] = -sc[0] / (float)NROWS;
}

extern "C" void kernel_launch(void* const* d_in, const int* in_sizes, int n_in,
                              void* d_out, int out_size, void* d_ws, size_t ws_size,
                              hipStream_t stream) {
    const float* feat   = (const float*)d_in[0];   // [2048, 2, 256] f32
    const int*   labels = (const int*)d_in[1];     // [2048, 1]
    float*       out    = (float*)d_out;           // scalar loss

    // workspace layout
    char* ws = (char*)d_ws;
    float* X       = (float*)ws;                               // 4 MB
    int*   cls     = (int*)(ws + (size_t)NROWS * DDIM * 4);    // 16 KB
    float* cnt     = (float*)((char*)cls + NROWS * 4);         // 512 B
    float* rowloss = (float*)((char*)cnt + NCLS_PAD * 4);      // 16 KB

    // 1) gather contrast layout (262144 float4 copies)
    build_contrast_kernel<<<1024, 256, 0, stream>>>(feat, X);
    // 2) class table + counts
    build_classes_kernel<<<1, 256, 0, stream>>>(labels, cls, cnt);
    // 3) fused WMMA strip GEMM + reductions
    //    dynamic LDS: 16*4096*4 + 4096*4 + 128*4 = 279,040 B  (< 320 KB WGP LDS)
    const size_t ldsBytes = (size_t)(STRIP_ROWS * NROWS + NROWS + NCLS_PAD) * 4;
    supcon_strip_kernel<<<NROWS / STRIP_ROWS, 256, ldsBytes, stream>>>(
        X, cls, cnt, rowloss);
    // 4) deterministic mean
    finalize_kernel<<<1, 256, 0, stream>>>(rowloss, out);
}